// Group_36919538876843
// MI455X (gfx1250) — compile-verified
//
#include <hip/hip_runtime.h>
#include <hip/hip_bf16.h>

#define B_ 16
#define N_ 16384
#define G_ 512
#define M_ 32

typedef float v2f __attribute__((ext_vector_type(2)));
typedef float v8f __attribute__((ext_vector_type(8)));

// ---------------------------------------------------------------------------
// Lexicographic compare-exchange across lanes (dist asc, idx asc tiebreak).
// takeMin: this lane keeps the smaller of {mine, partner}; else the larger.
// ---------------------------------------------------------------------------
__device__ __forceinline__ void cmpx(float& d, int& i, int mask, bool takeMin) {
    float od = __shfl_xor(d, mask, 32);
    int   oi = __shfl_xor(i, mask, 32);
    bool otherSmaller = (od < d) || (od == d && oi < i);
    if (takeMin == otherSmaller) { d = od; i = oi; }
}

// Async global->LDS copy of one dword per lane (gfx1250, ASYNCcnt-tracked).
// dsaddr = lds_off (VGPR) + imm ; mem = saddr (SGPR pair) + voff (VGPR) + imm
#define ASYNC_B32(lds_off, voff, gbase, imm)                                   \
    asm volatile("global_load_async_to_lds_b32 %0, %1, %2 offset:" #imm        \
                 :: "v"(lds_off), "v"(voff), "s"(gbase) : "memory")

// ===========================================================================
// Kernel 1: farthest point sampling, one workgroup per batch.
// Point cloud (SoA x/y/z, 192KB) lives in dynamic LDS (WGP has 320KB);
// the running min-distance array lives in 16 registers per thread
// (thread t owns n = t + k*1024, k ascending => first-index argmax kept).
// 511 dependent argmax rounds, all on-chip. Writes centers to d_out.
// ===========================================================================
__global__ void fps_kernel(const float* __restrict__ xyz,
                           float* __restrict__ center) {
    extern __shared__ float sm[];
    float* xs = sm;
    float* ys = sm + N_;
    float* zs = sm + 2 * N_;
    float* rv = sm + 3 * N_;               // 32 per-wave max values
    int*   ri = (int*)(sm + 3 * N_ + 32);  // 32 per-wave argmax indices
    int*   bc = (int*)(sm + 3 * N_ + 64);  // broadcast slot: current index

    const int b    = blockIdx.x;
    const int tid  = threadIdx.x;
    const int lane = tid & 31;
    const int wid  = tid >> 5;
    const float* base = xyz + (size_t)b * N_ * 3;

    float dloc[16];
    #pragma unroll
    for (int k = 0; k < 16; ++k) {
        int n = tid + (k << 10);
        float x = base[3 * n + 0], y = base[3 * n + 1], z = base[3 * n + 2];
        xs[n] = x; ys[n] = y; zs[n] = z;
        dloc[k] = 1e10f;
    }
    __syncthreads();
    if (tid == 0) {  // first center is point 0 (reference starts at index 0)
        center[(size_t)(b * G_) * 3 + 0] = xs[0];
        center[(size_t)(b * G_) * 3 + 1] = ys[0];
        center[(size_t)(b * G_) * 3 + 2] = zs[0];
        bc[0] = 0;
    }
    __syncthreads();

    for (int g = 1; g < G_; ++g) {
        const int cur = bc[0];
        const float lx = xs[cur], ly = ys[cur], lz = zs[cur];

        float bv = -1.0f; int bi = 0x7fffffff;
        #pragma unroll
        for (int k = 0; k < 16; ++k) {
            int n = tid + (k << 10);
            float dx = xs[n] - lx, dy = ys[n] - ly, dz = zs[n] - lz;
            float d  = dx * dx + dy * dy;   // match reference add order
            d += dz * dz;
            float nd = fminf(dloc[k], d);
            dloc[k] = nd;
            if (nd > bv) { bv = nd; bi = n; }  // strict > keeps first index
        }
        // wave32 butterfly argmax (larger value wins; tie -> smaller index)
        #pragma unroll
        for (int j = 16; j; j >>= 1) {
            float ov = __shfl_xor(bv, j, 32);
            int   oi = __shfl_xor(bi, j, 32);
            if (ov > bv || (ov == bv && oi < bi)) { bv = ov; bi = oi; }
        }
        if (lane == 0) { rv[wid] = bv; ri[wid] = bi; }
        __syncthreads();
        if (wid == 0) {
            float v = rv[lane]; int i2 = ri[lane];
            #pragma unroll
            for (int j = 16; j; j >>= 1) {
                float ov = __shfl_xor(v, j, 32);
                int   oi = __shfl_xor(i2, j, 32);
                if (ov > v || (ov == v && oi < i2)) { v = ov; i2 = oi; }
            }
            if (lane == 0) {
                bc[0] = i2;
                center[(size_t)(b * G_ + g) * 3 + 0] = xs[i2];
                center[(size_t)(b * G_ + g) * 3 + 1] = ys[i2];
                center[(size_t)(b * G_ + g) * 3 + 2] = zs[i2];
            }
        }
        __syncthreads();
    }
}

// ===========================================================================
// Kernel 2: kNN (exact top-32) + grouping. One workgroup = (batch, 16 centers).
// Distance tiles via V_WMMA_F32_16X16X4_F32:
//   A row m = [cx, cy, cz, |c|^2], B col n = [-2x, -2y, -2z, 1],
//   C[m][n] = |x_n|^2   ->   D = |c|^2 + |x|^2 - 2 c.x  (16x16 per WMMA).
// Point chunks are staged with double-buffered global_load_async_to_lds_b32
// (ASYNCcnt), overlapping the next chunk's fetch with WMMA + selection.
// Selection: wave w owns row w; resident top-32 sorted ascending across lanes
// in registers; a ballot early-exit skips batches that cannot beat the
// current 32nd-smallest (lex compare keeps top_k's stable tie order).
// ===========================================================================
__global__ void knn_group_kernel(const float* __restrict__ xyz,
                                 const float* __restrict__ center,
                                 float* __restrict__ groups) {
    __shared__ float cx[16], cy[16], cz[16], c2s[16];
    __shared__ float xs[512], ys[512], zs[512], n2[512]; // xs/ys/zs hold -2*coord
    __shared__ float dtile[16 * 512];     // 32 KB distance staging
    __shared__ float stage[2][512 * 3];   // 2 x 6 KB async landing buffers

    const int b    = blockIdx.y;
    const int g0   = blockIdx.x * 16;
    const int tid  = threadIdx.x;
    const int lane = tid & 31;
    const int w    = tid >> 5;            // 16 waves
    const int nlo  = lane & 15;
    const int hi   = lane >> 4;

    const float* pb = xyz + (size_t)b * N_ * 3;
    const unsigned voff = (unsigned)tid * 4u;   // per-lane dword offset

    // kick off async copy of chunk 0 (1536 dwords = 512 points)
    {
        unsigned l0 = (unsigned)(size_t)(&stage[0][0]) + voff;
        unsigned long long gb = (unsigned long long)(size_t)pb;
        ASYNC_B32(l0, voff, gb, 0);
        ASYNC_B32(l0, voff, gb, 2048);
        ASYNC_B32(l0, voff, gb, 4096);
    }

    if (tid < 16) {
        const float* c = center + (size_t)(b * G_ + g0 + tid) * 3;
        float x = c[0], y = c[1], z = c[2];
        cx[tid] = x; cy[tid] = y; cz[tid] = z;
        c2s[tid] = (x * x + y * y) + z * z;
    }
    __syncthreads();

    // A fragment (16x4 f32): lanes 0-15 hold K=0,1; lanes 16-31 hold K=2,3.
    // Branchless build (no exec-mask diamonds around the WMMA input).
    v2f afrag;
    afrag.x = hi ? cz[nlo] : cx[nlo];
    afrag.y = hi ? c2s[nlo] : cy[nlo];

    // resident top-32 (sorted ascending across lanes) + broadcast of lane 31
    float rd = __builtin_inff();
    int   rix = 0x7fffffff;
    float rmax = __builtin_inff();
    int   rmaxi = 0x7fffffff;

    for (int c = 0; c < 32; ++c) {              // 32 chunks of 512 points
        const int nb = c * 512;
        const int buf = c & 1;

        asm volatile("s_wait_asynccnt 0x0" ::: "memory"); // my async done
        __syncthreads();                                   // everyone's done

        // build SoA for this chunk: store -2x,-2y,-2z and |x|^2
        {
            float x = stage[buf][3 * tid + 0];
            float y = stage[buf][3 * tid + 1];
            float z = stage[buf][3 * tid + 2];
            xs[tid] = -2.0f * x; ys[tid] = -2.0f * y; zs[tid] = -2.0f * z;
            n2[tid] = (x * x + y * y) + z * z;
        }
        // overlap: start fetching chunk c+1 into the other buffer
        if (c < 31) {
            unsigned l1 = (unsigned)(size_t)(&stage[buf ^ 1][0]) + voff;
            unsigned long long gb =
                (unsigned long long)(size_t)(pb + 3 * (nb + 512));
            ASYNC_B32(l1, voff, gb, 0);
            ASYNC_B32(l1, voff, gb, 2048);
            ASYNC_B32(l1, voff, gb, 4096);
        }
        __syncthreads();

        // 32 column-tiles of 16; each wave computes 2 of them.
        #pragma unroll
        for (int t = 0; t < 2; ++t) {
            const int cb = (w + t * 16) * 16;
            const int n  = cb + nlo;
            v2f bfrag;                         // branchless: cndmask, no exec ops
            bfrag.x = hi ? zs[n] : xs[n];      // -2z | -2x
            bfrag.y = hi ? 1.0f  : ys[n];      //  1  | -2y
            v8f cfrag;
            float cc = n2[n];                  // |x_n|^2, N-index = lane&15
            #pragma unroll
            for (int r = 0; r < 8; ++r) cfrag[r] = cc;

            v8f dres = __builtin_amdgcn_wmma_f32_16x16x4_f32(
                false, afrag, false, bfrag, (short)0, cfrag, false, false);

            #pragma unroll
            for (int r = 0; r < 8; ++r)        // D: M = r + 8*hi, N = lane&15
                dtile[(r + hi * 8) * 512 + n] = dres[r];
        }
        __syncthreads();

        // wave w streams row w's 512 distances as 16 batches of 32
        for (int bb = 0; bb < 16; ++bb) {
            int   col = bb * 32 + lane;
            float cd  = dtile[w * 512 + col];
            int   ci  = nb + col;

            // early exit: can any candidate displace the resident 32nd?
            bool cand = (cd < rmax) || (cd == rmax && ci < rmaxi);
            if (__ballot(cand) == 0ull) continue;   // wave-uniform skip

            // bitonic sort ascending across 32 lanes
            #pragma unroll
            for (int k = 2; k <= 32; k <<= 1) {
                #pragma unroll
                for (int j = k >> 1; j; j >>= 1) {
                    bool asc = (lane & k) == 0;
                    bool low = (lane & j) == 0;
                    cmpx(cd, ci, j, low == asc);
                }
            }
            // reverse candidates, elementwise lex-min against resident set
            float crd = __shfl_xor(cd, 31, 32);
            int   cri = __shfl_xor(ci, 31, 32);
            bool cs = (crd < rd) || (crd == rd && cri < rix);
            if (cs) { rd = crd; rix = cri; }
            // resulting sequence is bitonic -> 5-stage ascending merge
            #pragma unroll
            for (int j = 16; j; j >>= 1) cmpx(rd, rix, j, (lane & j) == 0);
            // refresh the broadcast 32nd-smallest (lane 31 of resident)
            rmax  = __shfl(rd, 31, 32);
            rmaxi = __shfl(rix, 31, 32);
        }
    }

    // lane m of wave w holds the m-th nearest neighbor of row g0+w (sorted,
    // stable ties) -> gather and subtract center, matching reference order.
    {
        const int g = g0 + w;
        const int idx = rix;
        float px = pb[3 * idx + 0], py = pb[3 * idx + 1], pz = pb[3 * idx + 2];
        size_t o = (size_t)((b * G_ + g) * M_ + lane) * 3;
        groups[o + 0] = px - cx[w];
        groups[o + 1] = py - cy[w];
        groups[o + 2] = pz - cz[w];
    }
}

// ===========================================================================
extern "C" void kernel_launch(void* const* d_in, const int* in_sizes, int n_in,
                              void* d_out, int out_size, void* d_ws, size_t ws_size,
                              hipStream_t stream) {
    (void)in_sizes; (void)n_in; (void)out_size; (void)d_ws; (void)ws_size;
    const float* xyz = (const float*)d_in[0];
    float* groups = (float*)d_out;                             // [B,G,M,3]
    float* center = (float*)d_out + (size_t)B_ * G_ * M_ * 3;  // [B,G,3]

    // FPS: 16 blocks x 1024 threads, 192KB dynamic LDS (SoA coords)
    size_t smem = (size_t)(3 * N_ + 96) * sizeof(float);
    fps_kernel<<<B_, 1024, smem, stream>>>(xyz, center);

    // kNN + grouping: (G/16) x B workgroups, 512 threads (16 waves) each
    dim3 grid(G_ / 16, B_);
    knn_group_kernel<<<grid, 512, 0, stream>>>(xyz, center, groups);
}